// FKModel_39573828666218
// MI455X (gfx1250) — compile-verified
//
#include <hip/hip_runtime.h>
#include <hip/hip_bf16.h>

// Forward kinematics: B=262144 elements, J=24 joints, out (B, 75) fp32.
// Thread-per-element scalar chain; CDNA5 async LDS<->global data path.

#define NB   256
#define NJ   24
#define BATCH 262144
#define OUTF 75    // 3*(NJ+1)

#if __has_builtin(__builtin_amdgcn_global_load_async_to_lds_b128) && \
    __has_builtin(__builtin_amdgcn_global_store_async_from_lds_b128)
#define HAVE_ASYNC 1
#else
#define HAVE_ASYNC 0
#endif

typedef float v4f __attribute__((ext_vector_type(4)));
typedef int   v4i __attribute__((ext_vector_type(4)));

// Builtin signature (from clang diagnostic): param 1 is
// '__attribute__((vector_size(16))) int __device__ *'  i.e. v4i in addrspace(1).
#define AS_GLB_V4(p) ((__attribute__((address_space(1))) v4i*)(p))
#define AS_LDS_V4(p) ((__attribute__((address_space(3))) v4i*)(p))

__device__ __forceinline__ void g2l_16B(float* lds_dst, const float* glb_src) {
#if HAVE_ASYNC
    // async DMA: global -> LDS, 128 bits per lane, tracked by ASYNCcnt
    __builtin_amdgcn_global_load_async_to_lds_b128(AS_GLB_V4(glb_src), AS_LDS_V4(lds_dst), 0, 0);
#else
    *(v4f*)lds_dst = *(const v4f*)glb_src;
#endif
}

__device__ __forceinline__ void l2g_16B(float* glb_dst, const float* lds_src) {
#if HAVE_ASYNC
    __builtin_amdgcn_global_store_async_from_lds_b128(AS_GLB_V4(glb_dst), AS_LDS_V4(lds_src), 0, 0);
#else
    *(v4f*)glb_dst = *(const v4f*)lds_src;
#endif
}

__device__ __forceinline__ void wait_async0() {
#if HAVE_ASYNC
#if __has_builtin(__builtin_amdgcn_s_wait_asynccnt)
    __builtin_amdgcn_s_wait_asynccnt(0);
#else
    asm volatile("s_wait_asynccnt 0" ::: "memory");
#endif
#endif
}

__global__ __launch_bounds__(NB) void fk_kernel(
    const float* __restrict__ angles,   // (B, 24)
    const float* __restrict__ xyz,      // (24, 3)
    const float* __restrict__ rpy,      // (24, 3)
    const float* __restrict__ axis,     // (24, 3)
    float* __restrict__ out)            // (B, 75)
{
    // Per-joint constants: [0..8]=Ro row-major, [9..11]=xyz, [12..14]=axis, [15]=|axis|
    __shared__ __align__(16) float s_jc[NJ * 16];
    __shared__ __align__(16) float s_ang[NB * NJ];     // 24 KB angle tile
    __shared__ __align__(16) float s_out[NB * OUTF];   // 75 KB staging

    const int tid = threadIdx.x;

    // ---- Phase 1: kick off async load of this block's angle tile ----
    {
        const float* gsrc = angles + (size_t)blockIdx.x * (NB * NJ);
        #pragma unroll
        for (int i = 0; i < (NB * NJ) / (4 * NB); ++i) {   // 6 iterations
            int v = tid + i * NB;                          // 16B chunk index
            g2l_16B(&s_ang[v * 4], gsrc + v * 4);
        }
    }

    // ---- Phase 2 (overlapped): threads 0..23 build per-joint constants ----
    if (tid < NJ) {
        const int j = tid;
        float r = rpy[j * 3 + 0], p = rpy[j * 3 + 1], y = rpy[j * 3 + 2];
        float cr = __cosf(r), sr = __sinf(r);
        float cp = __cosf(p), sp = __sinf(p);
        float cy = __cosf(y), sy = __sinf(y);
        float* c = &s_jc[j * 16];
        c[0] = cy * cp; c[1] = cy * sp * sr - sy * cr; c[2] = cy * sp * cr + sy * sr;
        c[3] = sy * cp; c[4] = sy * sp * sr + cy * cr; c[5] = sy * sp * cr - cy * sr;
        c[6] = -sp;     c[7] = cp * sr;                c[8] = cp * cr;
        c[9]  = xyz[j * 3 + 0];
        c[10] = xyz[j * 3 + 1];
        c[11] = xyz[j * 3 + 2];
        float ax = axis[j * 3 + 0], ay = axis[j * 3 + 1], az = axis[j * 3 + 2];
        c[12] = ax; c[13] = ay; c[14] = az;
        c[15] = sqrtf(ax * ax + ay * ay + az * az);
    }

    wait_async0();       // this wave's async loads landed in LDS
    __syncthreads();     // all waves' tiles + constants visible

    // ---- Phase 3: per-element 24-joint chain, all in registers ----
    float p00 = 1.f, p01 = 0.f, p02 = 0.f;
    float p10 = 0.f, p11 = 1.f, p12 = 0.f;
    float p20 = 0.f, p21 = 0.f, p22 = 1.f;
    float ptx = 0.f, pty = 0.f, ptz = 0.f;

    float* ob = &s_out[tid * OUTF];          // stride 75 (odd) => conflict-free
    ob[0] = 0.f; ob[1] = 0.f; ob[2] = 0.f;   // base position

    const float* myang = &s_ang[tid * NJ];
    const v4f* jc4 = (const v4f*)s_jc;

    for (int j = 0; j < NJ; ++j) {
        const float th = myang[j];
        const v4f q0 = jc4[j * 4 + 0];   // o00 o01 o02 o10
        const v4f q1 = jc4[j * 4 + 1];   // o11 o12 o20 o21
        const v4f q2 = jc4[j * 4 + 2];   // o22 tx  ty  tz
        const v4f q3 = jc4[j * 4 + 3];   // ax  ay  az  len

        // Rodrigues: aa = axis*th, angle = |axis|*|th|, k = aa/max(angle,1e-6)
        const float ang = q3.w * fabsf(th);
        const float s   = th * (1.0f / fmaxf(ang, 1e-6f));
        const float kx = q3.x * s, ky = q3.y * s, kz = q3.z * s;
        const float ca = __cosf(ang), sa = __sinf(ang);
        const float oc = 1.0f - ca;
        const float r00 = fmaf(kx * kx, oc,  ca);
        const float r01 = fmaf(kx * ky, oc, -kz * sa);
        const float r02 = fmaf(kx * kz, oc,  ky * sa);
        const float r10 = fmaf(ky * kx, oc,  kz * sa);
        const float r11 = fmaf(ky * ky, oc,  ca);
        const float r12 = fmaf(ky * kz, oc, -kx * sa);
        const float r20 = fmaf(kz * kx, oc, -ky * sa);
        const float r21 = fmaf(kz * ky, oc,  kx * sa);
        const float r22 = fmaf(kz * kz, oc,  ca);

        // M = Ro * R   (3x3)
        const float o00 = q0.x, o01 = q0.y, o02 = q0.z, o10 = q0.w;
        const float o11 = q1.x, o12 = q1.y, o20 = q1.z, o21 = q1.w;
        const float o22 = q2.x;
        const float m00 = fmaf(o00, r00, fmaf(o01, r10, o02 * r20));
        const float m01 = fmaf(o00, r01, fmaf(o01, r11, o02 * r21));
        const float m02 = fmaf(o00, r02, fmaf(o01, r12, o02 * r22));
        const float m10 = fmaf(o10, r00, fmaf(o11, r10, o12 * r20));
        const float m11 = fmaf(o10, r01, fmaf(o11, r11, o12 * r21));
        const float m12 = fmaf(o10, r02, fmaf(o11, r12, o12 * r22));
        const float m20 = fmaf(o20, r00, fmaf(o21, r10, o22 * r20));
        const float m21 = fmaf(o20, r01, fmaf(o21, r11, o22 * r21));
        const float m22 = fmaf(o20, r02, fmaf(o21, r12, o22 * r22));

        // new P.R = P.R * M ; new P.t = P.R * xyz + P.t
        const float n00 = fmaf(p00, m00, fmaf(p01, m10, p02 * m20));
        const float n01 = fmaf(p00, m01, fmaf(p01, m11, p02 * m21));
        const float n02 = fmaf(p00, m02, fmaf(p01, m12, p02 * m22));
        const float n10 = fmaf(p10, m00, fmaf(p11, m10, p12 * m20));
        const float n11 = fmaf(p10, m01, fmaf(p11, m11, p12 * m21));
        const float n12 = fmaf(p10, m02, fmaf(p11, m12, p12 * m22));
        const float n20 = fmaf(p20, m00, fmaf(p21, m10, p22 * m20));
        const float n21 = fmaf(p20, m01, fmaf(p21, m11, p22 * m21));
        const float n22 = fmaf(p20, m02, fmaf(p21, m12, p22 * m22));
        const float tx = q2.y, ty = q2.z, tz = q2.w;
        const float ntx = fmaf(p00, tx, fmaf(p01, ty, fmaf(p02, tz, ptx)));
        const float nty = fmaf(p10, tx, fmaf(p11, ty, fmaf(p12, tz, pty)));
        const float ntz = fmaf(p20, tx, fmaf(p21, ty, fmaf(p22, tz, ptz)));

        p00 = n00; p01 = n01; p02 = n02;
        p10 = n10; p11 = n11; p12 = n12;
        p20 = n20; p21 = n21; p22 = n22;
        ptx = ntx; pty = nty; ptz = ntz;

        ob[3 * (j + 1) + 0] = ntx;
        ob[3 * (j + 1) + 1] = nty;
        ob[3 * (j + 1) + 2] = ntz;
    }

    __syncthreads();   // all staged outputs visible to all waves

    // ---- Phase 4: coalesced async store of contiguous 76.8 KB block tile ----
    {
        float* gdst = out + (size_t)blockIdx.x * (NB * OUTF);
        const int nvec = (NB * OUTF) / 4;   // 4800 x 16B chunks
        for (int v = tid; v < nvec; v += NB)
            l2g_16B(gdst + v * 4, &s_out[v * 4]);
        wait_async0();   // (s_endpgm also does an implicit wait-idle)
    }
}

extern "C" void kernel_launch(void* const* d_in, const int* in_sizes, int n_in,
                              void* d_out, int out_size, void* d_ws, size_t ws_size,
                              hipStream_t stream) {
    const float* angles = (const float*)d_in[0];   // (B, 24)
    const float* xyz    = (const float*)d_in[1];   // (24, 3)
    const float* rpy    = (const float*)d_in[2];   // (24, 3)
    const float* axis   = (const float*)d_in[3];   // (24, 3)
    float* out = (float*)d_out;                    // (B, 75)

    dim3 grid(BATCH / NB), block(NB);
    fk_kernel<<<grid, block, 0, stream>>>(angles, xyz, rpy, axis, out);
}